// AffineLog_52956946759704
// MI455X (gfx1250) — compile-verified
//
#include <hip/hip_runtime.h>

// CDNA5 / gfx1250: wave32, WMMA f32 16x16x4 used for the final projection GEMM.

typedef __attribute__((ext_vector_type(2))) float v2f;
typedef __attribute__((ext_vector_type(8))) float v8f;

#define K_SQRT   5
#define DB_ITERS 10
#define GREGORY  4
#define BLOCK    256

// 4x4 inverse via adjugate/cofactors (works on flat arrays, row- or col-major).
__device__ __forceinline__ void inv4(const double* __restrict__ m,
                                     double* __restrict__ o) {
    const double m0=m[0],  m1=m[1],  m2=m[2],  m3=m[3];
    const double m4=m[4],  m5=m[5],  m6=m[6],  m7=m[7];
    const double m8=m[8],  m9=m[9],  m10=m[10], m11=m[11];
    const double m12=m[12],m13=m[13],m14=m[14], m15=m[15];

    o[0]  =  m5*m10*m15 - m5*m11*m14 - m9*m6*m15 + m9*m7*m14 + m13*m6*m11 - m13*m7*m10;
    o[4]  = -m4*m10*m15 + m4*m11*m14 + m8*m6*m15 - m8*m7*m14 - m12*m6*m11 + m12*m7*m10;
    o[8]  =  m4*m9*m15  - m4*m11*m13 - m8*m5*m15 + m8*m7*m13 + m12*m5*m11 - m12*m7*m9;
    o[12] = -m4*m9*m14  + m4*m10*m13 + m8*m5*m14 - m8*m6*m13 - m12*m5*m10 + m12*m6*m9;
    o[1]  = -m1*m10*m15 + m1*m11*m14 + m9*m2*m15 - m9*m3*m14 - m13*m2*m11 + m13*m3*m10;
    o[5]  =  m0*m10*m15 - m0*m11*m14 - m8*m2*m15 + m8*m3*m14 + m12*m2*m11 - m12*m3*m10;
    o[9]  = -m0*m9*m15  + m0*m11*m13 + m8*m1*m15 - m8*m3*m13 - m12*m1*m11 + m12*m3*m9;
    o[13] =  m0*m9*m14  - m0*m10*m13 - m8*m1*m14 + m8*m2*m13 + m12*m1*m10 - m12*m2*m9;
    o[2]  =  m1*m6*m15  - m1*m7*m14  - m5*m2*m15 + m5*m3*m14 + m13*m2*m7  - m13*m3*m6;
    o[6]  = -m0*m6*m15  + m0*m7*m14  + m4*m2*m15 - m4*m3*m14 - m12*m2*m7  + m12*m3*m6;
    o[10] =  m0*m5*m15  - m0*m7*m13  - m4*m1*m15 + m4*m3*m13 + m12*m1*m7  - m12*m3*m5;
    o[14] = -m0*m5*m14  + m0*m6*m13  + m4*m1*m14 - m4*m2*m13 - m12*m1*m6  + m12*m2*m5;
    o[3]  = -m1*m6*m11  + m1*m7*m10  + m5*m2*m11 - m5*m3*m10 - m9*m2*m7   + m9*m3*m6;
    o[7]  =  m0*m6*m11  - m0*m7*m10  - m4*m2*m11 + m4*m3*m10 + m8*m2*m7   - m8*m3*m6;
    o[11] = -m0*m5*m11  + m0*m7*m9   + m4*m1*m11 - m4*m3*m9  - m8*m1*m7   + m8*m3*m5;
    o[15] =  m0*m5*m10  - m0*m6*m9   - m4*m1*m10 + m4*m2*m9  + m8*m1*m6   - m8*m2*m5;

    const double det = m0*o[0] + m1*o[4] + m2*o[8] + m3*o[12];
    const double rd  = 1.0 / det;
#pragma unroll
    for (int e = 0; e < 16; ++e) o[e] *= rd;
}

__device__ __forceinline__ void mm4(const double* __restrict__ a,
                                    const double* __restrict__ b,
                                    double* __restrict__ c) {
#pragma unroll
    for (int i = 0; i < 4; ++i)
#pragma unroll
        for (int j = 0; j < 4; ++j) {
            double s = a[i*4+0]*b[0*4+j];
            s = fma(a[i*4+1], b[1*4+j], s);
            s = fma(a[i*4+2], b[2*4+j], s);
            s = fma(a[i*4+3], b[3*4+j], s);
            c[i*4+j] = s;
        }
}

__global__ __launch_bounds__(BLOCK) void affine_log_proj_kernel(
    const float* __restrict__ A,       // (B,4,4) f32
    const float* __restrict__ basis,   // (7,4,4) f32
    float* __restrict__ out)           // (B,7)   f32
{
    // Padded stride 17 (gcd(17,64)=1) -> conflict-free column gathers.
    __shared__ float Ls[BLOCK * 17];

    const int tid = threadIdx.x;
    const int gb  = blockIdx.x * BLOCK + tid;

    // ---------------- per-lane f64 matrix log (matches reference numerics) ----
    double Y[16];
#pragma unroll
    for (int e = 0; e < 16; ++e) Y[e] = (double)A[gb * 16 + e];

    for (int s = 0; s < K_SQRT; ++s) {          // Y <- Y^(1/2) via Denman-Beavers
        double Z[16];
#pragma unroll
        for (int e = 0; e < 16; ++e) Z[e] = (e % 5 == 0) ? 1.0 : 0.0;
        for (int it = 0; it < DB_ITERS; ++it) {
            double iY[16], iZ[16];
            inv4(Y, iY);
            inv4(Z, iZ);
#pragma unroll
            for (int e = 0; e < 16; ++e) {
                const double yn = 0.5 * (Y[e] + iZ[e]);
                const double zn = 0.5 * (Z[e] + iY[e]);
                Y[e] = yn;
                Z[e] = zn;
            }
        }
    }

    // Gregory series: S = (X-I)(X+I)^-1 ; log = 2^(K+1) * sum S^(2k+1)/(2k+1)
    {
        double XmI[16], XpI[16];
#pragma unroll
        for (int e = 0; e < 16; ++e) {
            const double ident = (e % 5 == 0) ? 1.0 : 0.0;
            XmI[e] = Y[e] - ident;
            XpI[e] = Y[e] + ident;
        }
        double iXpI[16];
        inv4(XpI, iXpI);
        double S[16];
        mm4(XmI, iXpI, S);
        double S2[16];
        mm4(S, S, S2);
        double P[16], Lacc[16];
#pragma unroll
        for (int e = 0; e < 16; ++e) { P[e] = S[e]; Lacc[e] = S[e]; }
        for (int k = 1; k < GREGORY; ++k) {
            double T[16];
            mm4(P, S2, T);
            const double w = 1.0 / (double)(2 * k + 1);
#pragma unroll
            for (int e = 0; e < 16; ++e) {
                P[e] = T[e];
                Lacc[e] = fma(T[e], w, Lacc[e]);
            }
        }
        const double scale = (double)(1 << (K_SQRT + 1));   // 64
#pragma unroll
        for (int e = 0; e < 16; ++e)
            Ls[tid * 17 + e] = (float)(scale * Lacc[e]);
    }

    __syncthreads();

    // ---------------- projection GEMM via V_WMMA_F32_16X16X4_F32 --------------
    // out(16 rows x 7 cols) = L(16x16) x basis^T(16x16-padded), K chained 4x4.
    const int lane = tid & 31;
    const int wave = tid >> 5;
    const int h    = lane >> 4;      // lane half: K offset +2h in each fragment
    const int n    = lane & 15;      // M for A-frag, N for B/D-frags

    // B fragments: Bmat[K][N] = basis[N*16+K] for N<7, else 0 (zero-padded).
    v2f bfrag[4];
#pragma unroll
    for (int c = 0; c < 4; ++c) {
        const int k0 = 4 * c + 2 * h;
        float bx = 0.0f, by = 0.0f;
        if (n < 7) {
            bx = basis[n * 16 + k0];
            by = basis[n * 16 + k0 + 1];
        }
        bfrag[c].x = bx;
        bfrag[c].y = by;
    }

#pragma unroll
    for (int tile = 0; tile < 2; ++tile) {
        const int rowBase = wave * 32 + tile * 16;   // in-block row-0 of tile
        v8f acc = {};
#pragma unroll
        for (int c = 0; c < 4; ++c) {
            const int k0 = 4 * c + 2 * h;
            v2f a;
            a.x = Ls[(rowBase + n) * 17 + k0];
            a.y = Ls[(rowBase + n) * 17 + k0 + 1];
            // 8 args: (neg_a, A, neg_b, B, c_mod, C, reuse_a, reuse_b)
            acc = __builtin_amdgcn_wmma_f32_16x16x4_f32(
                false, a, false, bfrag[c], (short)0, acc, false, false);
        }
        const int gRow = blockIdx.x * BLOCK + rowBase;
#pragma unroll
        for (int r = 0; r < 8; ++r) {
            if (n < 7)
                out[(gRow + r + 8 * h) * 7 + n] = acc[r];
        }
    }
}

extern "C" void kernel_launch(void* const* d_in, const int* in_sizes, int n_in,
                              void* d_out, int out_size, void* d_ws, size_t ws_size,
                              hipStream_t stream) {
    const float* A     = (const float*)d_in[0];   // (65536,4,4) f32
    const float* basis = (const float*)d_in[1];   // (7,4,4) f32
    float* out         = (float*)d_out;           // (65536,7) f32

    const int batch  = in_sizes[0] / 16;          // 65536
    const int blocks = batch / BLOCK;             // 256

    affine_log_proj_kernel<<<dim3(blocks), dim3(BLOCK), 0, stream>>>(A, basis, out);
}